// PrototypicalNetwork_28484223108026
// MI455X (gfx1250) — compile-verified
//
#include <hip/hip_runtime.h>

// Prototypical network distances on gfx1250 (CDNA5, wave32).
// dist[q,c] = |q_emb|^2 + |p_c|^2 - 2 * (q_emb . p_c)
// q_emb = query @ W^T + b     (WMMA f32 16x16x4)
// p     = mean(support) @ W^T + b
//
// dist_kernel tiles 32 queries per block (2 M-subtiles) so every W (B-matrix)
// fragment feeds two WMMAs; all 8 wave32s get one output tile in phase 2.

typedef __attribute__((ext_vector_type(2))) float v2f;
typedef __attribute__((ext_vector_type(8))) float v8f;

#define DDIM 512
#define NC   64
#define KS   16
#define NQ   8192
#define MT   32          // queries per block in dist_kernel (2 x 16-row subtiles)
#define LDSP (DDIM + 4)  // padded LDS row: stride 516 mod 64 banks = 4 (conflict-free-ish)

// ---------------------------------------------------------------------------
// Kernel 1: class prototypes P[c][d'] = mean_k(support[c,k]) @ W^T + b,
//           plus pn[c] = |P[c]|^2.  64 blocks x 256 threads.
// ---------------------------------------------------------------------------
__global__ __launch_bounds__(256) void proto_kernel(
    const float* __restrict__ support,  // [NC*KS, D]
    const float* __restrict__ W,        // [D, D] (nn.Linear: out = x @ W^T)
    const float* __restrict__ b,        // [D]
    float* __restrict__ P,              // [NC, D]
    float* __restrict__ pn)             // [NC]
{
    __shared__ float smean[DDIM];
    __shared__ float red[256];

    const int c   = blockIdx.x;
    const int tid = threadIdx.x;

    // mean over the k_shot axis (affine projection commutes with the mean)
    for (int d = tid; d < DDIM; d += 256) {
        float acc = 0.0f;
        #pragma unroll
        for (int k = 0; k < KS; ++k)
            acc += support[(size_t)(c * KS + k) * DDIM + d];
        smean[d] = acc * (1.0f / (float)KS);
    }
    __syncthreads();

    // tiny GEMV: P[c, d'] = smean . W[d', :] + b[d']
    float sq = 0.0f;
    for (int dp = tid; dp < DDIM; dp += 256) {
        float acc = b[dp];
        const float* wr = W + (size_t)dp * DDIM;
        for (int d = 0; d < DDIM; ++d)
            acc = fmaf(smean[d], wr[d], acc);
        P[(size_t)c * DDIM + dp] = acc;
        sq = fmaf(acc, acc, sq);
    }

    red[tid] = sq;
    __syncthreads();
    for (int s = 128; s > 0; s >>= 1) {
        if (tid < s) red[tid] += red[tid + s];
        __syncthreads();
    }
    if (tid == 0) pn[c] = red[0];
}

// ---------------------------------------------------------------------------
// Kernel 2: per block, 32 queries:
//   phase 1: q_emb[32,512] = query_tile @ W^T + b   (WMMA, kept in LDS)
//   phase 2: cross[32,64]  = q_emb @ P^T            (WMMA, 1 tile per wave)
//   epilogue: out = qn + pn - 2*cross
// 256 threads = 8 wave32s.
// ---------------------------------------------------------------------------
__global__ __launch_bounds__(256) void dist_kernel(
    const float* __restrict__ query,    // [NQ, D]
    const float* __restrict__ W,        // [D, D]
    const float* __restrict__ b,        // [D]
    const float* __restrict__ P,        // [NC, D]
    const float* __restrict__ pn,       // [NC]
    float* __restrict__ out)            // [NQ, NC]
{
    __shared__ float emb[MT][LDSP];     // q_emb tile (~66 KB; CDNA5 WGP has 320 KB LDS)
    __shared__ float qn[MT];            // |q_emb|^2 per row

    const int tid  = threadIdx.x;
    const int wave = tid >> 5;          // 0..7
    const int lane = tid & 31;
    const int h    = lane >> 4;         // which 16-lane half
    const int l16  = lane & 15;
    const int qb   = blockIdx.x * MT;   // query tile base

    if (tid < MT) qn[tid] = 0.0f;

    // ---------------- phase 1: q_emb = query @ W^T + b -------------------
    // 32 N-tiles of 16 output dims; wave w owns n-tiles {w, w+8, w+16, w+24}
    // and BOTH 16-row M-subtiles, so each B fragment is reused twice.
    {
        v8f acc[2][4] = {};
        const int nb0 = wave * 16;
        for (int kb = 0; kb < DDIM; kb += 4) {
            const int ka = kb + 2 * h;  // this half's K pair
            // A fragments: A[m = l16 (+16)][k = ka, ka+1]  (16x4 f32, 2 VGPRs)
            v2f a0 = *(const v2f*)(query + (size_t)(qb + l16) * DDIM + ka);
            v2f a1 = *(const v2f*)(query + (size_t)(qb + 16 + l16) * DDIM + ka);
            #pragma unroll
            for (int j = 0; j < 4; ++j) {
                const int nb = nb0 + j * 128;  // (wave + 8j) * 16
                // B fragment: B[k][n] = W[n][k], n = nb + l16
                v2f bv = *(const v2f*)(W + (size_t)(nb + l16) * DDIM + ka);
                acc[0][j] = __builtin_amdgcn_wmma_f32_16x16x4_f32(
                    false, a0, false, bv, (short)0, acc[0][j], false, false);
                acc[1][j] = __builtin_amdgcn_wmma_f32_16x16x4_f32(
                    false, a1, false, bv, (short)0, acc[1][j], false, false);
            }
        }
        // store C tiles to LDS with bias: element (M = v + 8h, N = l16)
        #pragma unroll
        for (int mi = 0; mi < 2; ++mi) {
            #pragma unroll
            for (int j = 0; j < 4; ++j) {
                const int nb   = nb0 + j * 128;
                const float bv = b[nb + l16];
                #pragma unroll
                for (int v = 0; v < 8; ++v)
                    emb[mi * 16 + v + 8 * h][nb + l16] = acc[mi][j][v] + bv;
            }
        }
    }
    __syncthreads();

    // ---------------- |q_emb|^2 reduction (ds_add_f32) --------------------
    {
        const int r  = tid & (MT - 1);  // row 0..31
        const int ch = tid >> 5;        // chunk 0..7, 64 elems each
        float s = 0.0f;
        #pragma unroll
        for (int i = 0; i < 64; ++i) {
            float e = emb[r][ch * 64 + i];
            s = fmaf(e, e, s);
        }
        atomicAdd(&qn[r], s);
    }
    __syncthreads();

    // ---------------- phase 2: cross = q_emb @ P^T, fused epilogue --------
    // 8 output tiles (32 queries x 64 classes = 2 x 4); one per wave.
    {
        const int mi = wave >> 2;       // M-subtile 0..1
        const int nb = (wave & 3) * 16; // class tile base
        v8f acc = {};
        for (int kb = 0; kb < DDIM; kb += 4) {
            const int ka = kb + 2 * h;
            // A from LDS: A[m = l16][k]
            v2f a = *(const v2f*)&emb[mi * 16 + l16][ka];
            // B fragment: B[k][n] = P[class = nb + l16][k]
            v2f bv = *(const v2f*)(P + (size_t)(nb + l16) * DDIM + ka);
            acc = __builtin_amdgcn_wmma_f32_16x16x4_f32(
                false, a, false, bv, (short)0, acc, false, false);
        }
        const float pnv = pn[nb + l16];
        #pragma unroll
        for (int v = 0; v < 8; ++v) {
            const int m = mi * 16 + v + 8 * h;    // query row in tile
            const int c = nb + l16;               // class
            out[(size_t)(qb + m) * NC + c] = qn[m] + pnv - 2.0f * acc[v];
        }
    }
}

// ---------------------------------------------------------------------------
extern "C" void kernel_launch(void* const* d_in, const int* in_sizes, int n_in,
                              void* d_out, int out_size, void* d_ws, size_t ws_size,
                              hipStream_t stream) {
    const float* support = (const float*)d_in[0];   // [NC*KS, D]
    const float* query   = (const float*)d_in[1];   // [NQ, D]
    const float* W       = (const float*)d_in[2];   // [D, D]
    const float* b       = (const float*)d_in[3];   // [D]
    // d_in[4], d_in[5]: n_way / k_shot scalars (compile-time constants here)

    float* P  = (float*)d_ws;                       // [NC, D]
    float* pn = P + (size_t)NC * DDIM;              // [NC]
    float* out = (float*)d_out;                     // [NQ, NC]

    proto_kernel<<<NC, 256, 0, stream>>>(support, W, b, P, pn);
    dist_kernel<<<NQ / MT, 256, 0, stream>>>(query, W, b, P, pn, out);
}